// RecursiveRotorAccumulator_62113817034790
// MI455X (gfx1250) — compile-verified
//
#include <hip/hip_runtime.h>
#include <math.h>

typedef __attribute__((ext_vector_type(2))) float v2f;
typedef __attribute__((ext_vector_type(8))) float v8f;

// ---------------- Compile-time Cayley sign table (Cl(4,1): SIG = [1,1,1,1,-1]) ---
constexpr int pc_(unsigned v) { int c = 0; while (v) { c += (v & 1); v >>= 1; } return c; }
constexpr float gasign_(int a, int b) {
    int sw = 0;
    int t = a >> 1;
    while (t) { sw += pc_(unsigned(t & b)); t >>= 1; }
    float s = (sw & 1) ? -1.0f : 1.0f;
    if ((a & b) & 16) s = -s;   // only generator e5 has SIG = -1
    return s;
}
struct SignTab { float v[32][32]; };
constexpr SignTab buildSigns_() {
    SignTab t{};
    for (int a = 0; a < 32; ++a)
        for (int k = 0; k < 32; ++k)
            t.v[a][k] = gasign_(a, a ^ k);   // e_a * e_{a^k} -> blade k
    return t;
}
static constexpr SignTab SGN = buildSigns_();

// ---------------- geometric product + manifold normalization (per-thread) -------
// out[k] = sum_a SGN[a][k] * right[a] * left[a^k]   (signs folded at compile time)
__device__ __forceinline__ void gp_norm(const float* __restrict__ r,
                                        const float* __restrict__ l,
                                        float* __restrict__ o) {
#pragma unroll
    for (int k = 0; k < 32; ++k) {
        float acc = 0.0f;
#pragma unroll
        for (int a = 0; a < 32; ++a)
            acc = fmaf(SGN.v[a][k] * r[a], l[a ^ k], acc);
        o[k] = acc;
    }
    float q = 0.0f;
#pragma unroll
    for (int g = 0; g < 32; ++g) {
        float p = o[g] * o[g];
        q += (g & 16) ? -p : p;          // metric[g] = -1 iff bit4 set
    }
    float sc = rsqrtf(fabsf(q) + 1e-6f);
#pragma unroll
    for (int g = 0; g < 32; ++g) o[g] *= sc;
}

// ---------------- Phase 1: delta = x@W + b via V_WMMA_F32_16X16X4_F32 ----------
// One wave per (b,s) site. delta[e,g] = sum_d W[d,e] * x[d,g]  (A = W^T, B = x)
__global__ void rotor_init_kernel(const float* __restrict__ x,
                                  const float* __restrict__ W,
                                  const float* __restrict__ bias,
                                  float* __restrict__ out, int nsites) {
    int wave = (blockIdx.x * blockDim.x + threadIdx.x) >> 5;
    int lane = threadIdx.x & 31;
    if (wave >= nsites) return;          // wave-uniform: EXEC stays all-ones

    const int m  = lane & 15;            // A: row M=e ; B/C: col N
    const int hi = lane >> 4;            // K / M-half selector

    // A tiles (16x4 f32): lane holds A[m, 2*hi] , A[m, 2*hi+1]; A[e,k]=W[d0+k, e]
    v2f A[4];
#pragma unroll
    for (int c = 0; c < 4; ++c) {
        int d0 = 4 * c + 2 * hi;
        A[c].x = W[d0 * 16 + m];
        A[c].y = W[(d0 + 1) * 16 + m];
    }

    const float* xb = x + (size_t)wave * 512;   // 16 d * 32 g
    v8f acc0 = {};   // g = 0..15
    v8f acc1 = {};   // g = 16..31
#pragma unroll
    for (int c = 0; c < 4; ++c) {
        int d0 = 4 * c + 2 * hi;
        v2f B0, B1;
        B0.x = xb[d0 * 32 + m];          B0.y = xb[(d0 + 1) * 32 + m];
        B1.x = xb[d0 * 32 + 16 + m];     B1.y = xb[(d0 + 1) * 32 + 16 + m];
        acc0 = __builtin_amdgcn_wmma_f32_16x16x4_f32(false, A[c], false, B0,
                                                     (short)0, acc0, false, false);
        acc1 = __builtin_amdgcn_wmma_f32_16x16x4_f32(false, A[c], false, B1,
                                                     (short)0, acc1, false, false);
    }

    // rotors = 0.5*(delta + bias); rotors[...,0] += 1
    float bv0 = bias[m];
    float bv1 = bias[16 + m];
    float r0[8], r1[8];
#pragma unroll
    for (int v = 0; v < 8; ++v) {
        r0[v] = 0.5f * (acc0[v] + bv0);
        r1[v] = 0.5f * (acc1[v] + bv1);
    }
    if (m == 0) {                        // g==0 lives at lanes 0 and 16
#pragma unroll
        for (int v = 0; v < 8; ++v) r0[v] += 1.0f;
    }

    // Per-e normalization: e = v + 8*hi spans 16 lanes (N) x two g-halves.
    // metric: +1 for g<16 half, -1 for g>=16 half. Butterfly within 16-lane group.
#pragma unroll
    for (int v = 0; v < 8; ++v) {
        float p = r0[v] * r0[v] - r1[v] * r1[v];
        p += __shfl_xor(p, 1, 32);
        p += __shfl_xor(p, 2, 32);
        p += __shfl_xor(p, 4, 32);
        p += __shfl_xor(p, 8, 32);
        float sc = rsqrtf(fabsf(p) + 1e-6f);
        r0[v] *= sc;
        r1[v] *= sc;
    }

    float* ob = out + (size_t)wave * 512;
#pragma unroll
    for (int v = 0; v < 8; ++v) {
        int e = v + 8 * hi;
        ob[e * 32 + m]      = r0[v];
        ob[e * 32 + 16 + m] = r1[v];
    }
}

// ---------------- Phase 2: one Hillis-Steele doubling step ---------------------
// new[s] = s < step ? old[s] : normalize(gp(old[s], old[s-step]))
// One thread per multivector; layout index = ((b*4096 + s)*16 + e), g innermost.
__global__ void rotor_scan_kernel(const float* __restrict__ in,
                                  float* __restrict__ out, int step) {
    int idx = blockIdx.x * blockDim.x + threadIdx.x;   // 0 .. 262143
    int s = (idx >> 4) & 4095;

    const float4* rp = (const float4*)(in + (size_t)idx * 32);
    float R[32];
#pragma unroll
    for (int i = 0; i < 8; ++i) {
        float4 t = rp[i];
        R[4 * i + 0] = t.x; R[4 * i + 1] = t.y;
        R[4 * i + 2] = t.z; R[4 * i + 3] = t.w;
    }

    float4* op = (float4*)(out + (size_t)idx * 32);
    if (s < step) {                       // passthrough prefix
#pragma unroll
        for (int i = 0; i < 8; ++i) op[i] = rp[i];
        return;
    }

    const float4* lp = (const float4*)(in + ((size_t)idx - (size_t)step * 16) * 32);
    float L[32];
#pragma unroll
    for (int i = 0; i < 8; ++i) {
        float4 t = lp[i];
        L[4 * i + 0] = t.x; L[4 * i + 1] = t.y;
        L[4 * i + 2] = t.z; L[4 * i + 3] = t.w;
    }

    float O[32];
    gp_norm(R, L, O);                     // right = this element, left = s-step

#pragma unroll
    for (int i = 0; i < 8; ++i) {
        float4 t;
        t.x = O[4 * i + 0]; t.y = O[4 * i + 1];
        t.z = O[4 * i + 2]; t.w = O[4 * i + 3];
        op[i] = t;
    }
}

// ---------------- host glue ----------------------------------------------------
extern "C" void kernel_launch(void* const* d_in, const int* in_sizes, int n_in,
                              void* d_out, int out_size, void* d_ws, size_t ws_size,
                              hipStream_t stream) {
    const float* x = (const float*)d_in[0];   // (4,4096,16,32)
    const float* W = (const float*)d_in[1];   // (16,16)
    const float* b = (const float*)d_in[2];   // (32,)

    float* bufA = (float*)d_out;              // final result must land here
    float* bufB = (float*)d_ws;               // 32 MB scratch

    const int nsites = 4 * 4096;              // B*S waves, one wave per site
    rotor_init_kernel<<<nsites / 8, 256, 0, stream>>>(x, W, b, bufA, nsites);

    // 12 doubling steps, ping-pong A->B->A...; even count => final in d_out
    const int nmv = 4 * 4096 * 16;            // 262144 multivectors
    float* src = bufA;
    float* dst = bufB;
    for (int step = 1; step < 4096; step <<= 1) {
        rotor_scan_kernel<<<nmv / 256, 256, 0, stream>>>(src, dst, step);
        float* t = src; src = dst; dst = t;
    }
}